// CovQuadraticCrossEntropyLoss_17025250361380
// MI455X (gfx1250) — compile-verified
//
#include <hip/hip_runtime.h>
#include <hip/hip_bf16.h>

typedef __attribute__((ext_vector_type(2))) float v2f;
typedef __attribute__((ext_vector_type(8))) float v8f;

#define V_DIM 512

#if defined(__HIP_DEVICE_COMPILE__)
#if __has_builtin(__builtin_amdgcn_wmma_f32_16x16x4_f32)
#define USE_WMMA_F32 1
#endif
#endif

// One block per (s,b). 512 threads = 16 wave32.
// out[sb] = logsumexp(m) - m[target] + 0.5*(sum_i K_ii p_i - p^T K p)
__global__ __launch_bounds__(512) void ccel_kernel(
    const float* __restrict__ m,      // [SB, V]
    const float* __restrict__ K,      // [SB, V, V]
    const int* __restrict__ tgt,      // [SB]
    float* __restrict__ out)          // [SB]
{
    const int sb = blockIdx.x;
    const int t  = threadIdx.x;              // 0..511
    const float* mrow = m + (size_t)sb * V_DIM;
    const float* Kmat = K + (size_t)sb * V_DIM * V_DIM;

    __shared__ __align__(16) float p_lds[V_DIM];
    __shared__ __align__(16) float red[V_DIM];

    // ---- softmax over V=512 (one element per thread) ----
    const float x = mrow[t];
    red[t] = x; __syncthreads();
    for (int s = V_DIM / 2; s > 0; s >>= 1) {
        if (t < s) red[t] = fmaxf(red[t], red[t + s]);
        __syncthreads();
    }
    const float mx = red[0];
    __syncthreads();

    const float e = __expf(x - mx);
    red[t] = e; __syncthreads();
    for (int s = V_DIM / 2; s > 0; s >>= 1) {
        if (t < s) red[t] += red[t + s];
        __syncthreads();
    }
    const float sumexp = red[0];
    __syncthreads();

    const float p = e / sumexp;
    p_lds[t] = p;

    // ---- diag term: sum_i K_ii * p_i ----
    const float dg = Kmat[(size_t)t * V_DIM + t] * p;
    red[t] = dg; __syncthreads();
    for (int s = V_DIM / 2; s > 0; s >>= 1) {
        if (t < s) red[t] += red[t + s];
        __syncthreads();
    }
    const float diag = red[0];
    __syncthreads();   // p_lds fully populated + red free for reuse

    // ---- quadratic form p^T K p (streams the 1 MB K matrix once) ----
    float qp;
    float qscale;
#if defined(USE_WMMA_F32)
    {
        const int lane = t & 31;
        const int wave = t >> 5;               // 0..15
        const int kk   = (lane >> 4) << 1;     // 0 for lanes<16, 2 for lanes>=16
        qp = 0.0f;
        // Each wave handles i-blocks {wave, wave+16}; full j sweep each.
        for (int rep = 0; rep < 2; ++rep) {
            const int i0  = (wave + rep * 16) * 16;
            const int row = i0 + (lane & 15);
            const float* arow = Kmat + (size_t)row * V_DIM + kk;
            v8f acc = {};
            #pragma unroll 8
            for (int j0 = 0; j0 < V_DIM; j0 += 4) {
                // A tile 16x4 f32: lane (M=lane&15) supplies K[row, j0+kk], K[row, j0+kk+1]
                v2f a = __builtin_nontemporal_load((const v2f*)(arow + j0));
                // B tile 4x16 f32: p[j0+kk], p[j0+kk+1] broadcast across all N columns
                v2f b = *(const v2f*)&p_lds[j0 + kk];
                acc = __builtin_amdgcn_wmma_f32_16x16x4_f32(
                    /*neg_a=*/false, a, /*neg_b=*/false, b,
                    /*c_mod=*/(short)0, acc, /*reuse_a=*/false, /*reuse_b=*/false);
            }
            // D[i, c] == (K_block @ p)_i replicated in every column c.
            // Lane holds D[M = r + 8*(lane>>4), N = lane&15] in acc[r].
            const int mbase = i0 + ((lane >> 4) << 3);
            #pragma unroll
            for (int r = 0; r < 8; ++r)
                qp += acc[r] * p_lds[mbase + r];
        }
        // Block-sum of qp counts every column => 16x the true value.
        qscale = 1.0f / 16.0f;
    }
#else
    {
        // Fallback: per-thread GEMV row (still correct f32; no WMMA emitted).
        const float* arow = Kmat + (size_t)t * V_DIM;
        float kp = 0.0f;
        #pragma unroll 8
        for (int j = 0; j < V_DIM; ++j)
            kp += __builtin_nontemporal_load(arow + j) * p_lds[j];
        qp = kp * p;
        qscale = 1.0f;
    }
#endif

    red[t] = qp; __syncthreads();
    for (int s = V_DIM / 2; s > 0; s >>= 1) {
        if (t < s) red[t] += red[t + s];
        __syncthreads();
    }
    const float pKp = red[0] * qscale;

    if (t == 0) {
        const float lse = mx + __logf(sumexp);
        const float ce  = lse - mrow[tgt[sb]];
        out[sb] = ce + 0.5f * (diag - pKp);
    }
}

extern "C" void kernel_launch(void* const* d_in, const int* in_sizes, int n_in,
                              void* d_out, int out_size, void* d_ws, size_t ws_size,
                              hipStream_t stream) {
    (void)in_sizes; (void)n_in; (void)d_ws; (void)ws_size;
    const float* m   = (const float*)d_in[0];   // [32,16,512] f32
    const float* K   = (const float*)d_in[1];   // [32,16,512,512] f32
    const int*   tgt = (const int*)d_in[2];     // [32,16] i32
    float* out = (float*)d_out;                 // [32,16] f32 -> out_size = 512
    const int sb = out_size;                    // 512 (s,b) pairs
    ccel_kernel<<<sb, V_DIM, 0, stream>>>(m, K, tgt, out);
}